// GATModel_9483287789909
// MI455X (gfx1250) — compile-verified
//
#include <hip/hip_runtime.h>
#include <hip/hip_bf16.h>

#define NEG_SLOPE 0.2f
#define BN_EPS    1e-5f

typedef float v2f __attribute__((ext_vector_type(2)));
typedef float v8f __attribute__((ext_vector_type(8)));

__device__ __forceinline__ float lrelu(float x) { return x > 0.f ? x : NEG_SLOPE * x; }

// Order-preserving float<->uint encoding so float-max can use integer atomicMax.
__device__ __forceinline__ unsigned fenc(float x) {
    unsigned u = __float_as_uint(x);
    return (u & 0x80000000u) ? ~u : (u | 0x80000000u);
}
__device__ __forceinline__ float fdec(unsigned u) {
    return (u & 0x80000000u) ? __uint_as_float(u ^ 0x80000000u) : __uint_as_float(~u);
}
#define ENC_NEG_INF 0x007FFFFFu  // fenc(-inf)

// ---------------------------------------------------------------------------
// fills
// ---------------------------------------------------------------------------
__global__ void fill_f32(float* __restrict__ p, size_t n, float v) {
    size_t i = (size_t)blockIdx.x * 256 + threadIdx.x;
    if (i < n) p[i] = v;
}
__global__ void fill_u32(unsigned* __restrict__ p, size_t n, unsigned v) {
    size_t i = (size_t)blockIdx.x * 256 + threadIdx.x;
    if (i < n) p[i] = v;
}

// ---------------------------------------------------------------------------
// C[M x Nw] = A[M x K] @ W[K x Nw] + bias  via  v_wmma_f32_16x16x4_f32
//
// Block = 128 threads (4 waves), computes a 64(M) x 64(N) tile.
// K is consumed in 32-wide panels staged in LDS:
//   As[64 rows][36]  : A panel, row-major, stride 36 (16B-aligned stores,
//                      36*row mod 64 = 4*(9*row mod 16) -> spread banks)
//   Bs[64 cols][36]  : W panel stored K-contiguous (transposed) so a B
//                      fragment is one b64, symmetric with A.
// Each wave owns one 16-col subtile and 4 stacked 16-row subtiles:
//   per K-step of 4: 1 b64 B-frag + 4 b64 A-frags -> 4 WMMAs (B reused).
// WMMA fragment mapping (ISA 7.12.2, wave32):
//   A 16x4 : lanes 0-15 k={kb,kb+1}, lanes 16-31 k={kb+2,kb+3}
//   B 4x16 : VGPR0 k=kb(+2*half), VGPR1 k=kb+1(+2*half), n = lane&15
//   C 16x16: VGPR r -> row r (lanes 0-15) / r+8 (lanes 16-31)
// Requires Nw%64==0, K%32==0. M tail handled by clamped loads (EXEC stays
// all-ones for WMMA) and predicated stores.
// ---------------------------------------------------------------------------
#define KB 32
#define LSTR 36

__global__ __launch_bounds__(128) void gemm_wmma_f32(
    const float* __restrict__ A, const float* __restrict__ W,
    const float* __restrict__ bias, float* __restrict__ C,
    int M, int K, int Nw)
{
    __shared__ float As[64 * LSTR];
    __shared__ float Bs[64 * LSTR];

    const int tid  = threadIdx.x;
    const int wave = tid >> 5;
    const int lane = tid & 31;
    const int hh   = lane >> 4;     // half-wave selects K pair
    const int l16  = lane & 15;

    const int m0 = blockIdx.y * 64;
    const int n0 = blockIdx.x * 64;
    const int nw = n0 + wave * 16;  // this wave's 16 output columns

    v8f acc[4] = {v8f{}, v8f{}, v8f{}, v8f{}};

    for (int kb0 = 0; kb0 < K; kb0 += KB) {
        // ---- cooperative staging: 2 x 2048 floats, 4 float4 per thread each
        #pragma unroll
        for (int i = 0; i < 4; ++i) {
            int idx4 = tid + 128 * i;              // 0..511
            // A panel: 64 rows x 32 k   (8 float4 per row)
            int row  = idx4 >> 3;
            int k4   = (idx4 & 7) * 4;
            int grow = m0 + row; if (grow >= M) grow = M - 1;
            const float* gA = A + (size_t)grow * K + kb0 + k4;
            *(float4*)(&As[row * LSTR + k4]) = *(const float4*)gA;
            if (kb0 + KB < K) __builtin_prefetch(gA + KB, 0, 1);
            // W panel: 32 k x 64 n      (16 float4 per k-row), store transposed
            int krow = idx4 >> 4;
            int n4   = (idx4 & 15) * 4;
            const float* gW = W + (size_t)(kb0 + krow) * Nw + n0 + n4;
            float4 vB = *(const float4*)gW;
            if (kb0 + KB < K) __builtin_prefetch(gW + (size_t)KB * Nw, 0, 1);
            Bs[(n4 + 0) * LSTR + krow] = vB.x;
            Bs[(n4 + 1) * LSTR + krow] = vB.y;
            Bs[(n4 + 2) * LSTR + krow] = vB.z;
            Bs[(n4 + 3) * LSTR + krow] = vB.w;
        }
        __syncthreads();

        const float* pa = As + l16 * LSTR + 2 * hh;
        const float* pb = Bs + (wave * 16 + l16) * LSTR + 2 * hh;
        #pragma unroll
        for (int kk = 0; kk < KB; kk += 4) {
            v2f b  = *(const v2f*)(pb + kk);
            v2f a0 = *(const v2f*)(pa + kk);
            v2f a1 = *(const v2f*)(pa + 16 * LSTR + kk);
            v2f a2 = *(const v2f*)(pa + 32 * LSTR + kk);
            v2f a3 = *(const v2f*)(pa + 48 * LSTR + kk);
            acc[0] = __builtin_amdgcn_wmma_f32_16x16x4_f32(false, a0, false, b, (short)0, acc[0], false, false);
            acc[1] = __builtin_amdgcn_wmma_f32_16x16x4_f32(false, a1, false, b, (short)0, acc[1], false, false);
            acc[2] = __builtin_amdgcn_wmma_f32_16x16x4_f32(false, a2, false, b, (short)0, acc[2], false, false);
            acc[3] = __builtin_amdgcn_wmma_f32_16x16x4_f32(false, a3, false, b, (short)0, acc[3], false, false);
        }
        __syncthreads();
    }

    float bv = bias[nw + l16];
    #pragma unroll
    for (int ms = 0; ms < 4; ++ms) {
        #pragma unroll
        for (int r = 0; r < 8; ++r) {
            int row = m0 + ms * 16 + r + 8 * hh;
            if (row < M) C[(size_t)row * Nw + nw + l16] = acc[ms][r] + bv;
        }
    }
}

// ---------------------------------------------------------------------------
// Layer-1 edge kernels (H=4, D=64, feature width 256)
// ---------------------------------------------------------------------------
__global__ __launch_bounds__(256) void gat_e1_logits(
    const float* __restrict__ xl, const float* __restrict__ xr,
    const float* __restrict__ att, const int* __restrict__ ei,
    int E, int Etot, float* __restrict__ logit, unsigned* __restrict__ mx)
{
    int e = blockIdx.x * 8 + (threadIdx.x >> 5);     // one wave per edge
    if (e >= Etot) return;
    int lane = threadIdx.x & 31;
    int h  = lane >> 3;                              // 8 lanes per head
    int d0 = (lane & 7) * 8;
    int src, dst;
    if (e < E) { src = ei[e]; dst = ei[E + e]; } else { src = dst = e - E; }

    const float* pl = xl + (size_t)src * 256 + h * 64 + d0;
    const float* pr = xr + (size_t)dst * 256 + h * 64 + d0;
    const float* pa = att + h * 64 + d0;
    float s = 0.f;
    #pragma unroll
    for (int j = 0; j < 8; j += 4) {
        float4 a = *(const float4*)(pa + j);
        float4 l = *(const float4*)(pl + j);
        float4 r = *(const float4*)(pr + j);
        s += a.x * lrelu(l.x + r.x) + a.y * lrelu(l.y + r.y)
           + a.z * lrelu(l.z + r.z) + a.w * lrelu(l.w + r.w);
    }
    s += __shfl_xor(s, 4); s += __shfl_xor(s, 2); s += __shfl_xor(s, 1);
    if ((lane & 7) == 0) {
        logit[(size_t)e * 4 + h] = s;
        atomicMax(&mx[(size_t)dst * 4 + h], fenc(s));
    }
}

__global__ __launch_bounds__(256) void gat_e1_accum(
    const float* __restrict__ xl, const float* __restrict__ logit,
    const unsigned* __restrict__ mx, const int* __restrict__ ei,
    int E, int Etot, float* __restrict__ den, float* __restrict__ acc)
{
    int e = blockIdx.x * 4 + (threadIdx.x >> 6);     // 64 threads per edge
    if (e >= Etot) return;
    int t = threadIdx.x & 63;
    int h = t >> 4;
    int i = t * 4;                                   // covers 256 dims
    int src, dst;
    if (e < E) { src = ei[e]; dst = ei[E + e]; } else { src = dst = e - E; }

    float a = __expf(logit[(size_t)e * 4 + h] - fdec(mx[(size_t)dst * 4 + h]));
    if ((t & 15) == 0) atomicAdd(&den[(size_t)dst * 4 + h], a);
    float4 xv = *(const float4*)(xl + (size_t)src * 256 + i);
    float* po = acc + (size_t)dst * 256 + i;
    atomicAdd(po + 0, a * xv.x);
    atomicAdd(po + 1, a * xv.y);
    atomicAdd(po + 2, a * xv.z);
    atomicAdd(po + 3, a * xv.w);
}

__global__ void gat_n1_post(
    float* __restrict__ buf, const float* __restrict__ den,
    const float* __restrict__ bias, const float* __restrict__ g,
    const float* __restrict__ b, const float* __restrict__ m,
    const float* __restrict__ v, int N)
{
    size_t idx = (size_t)blockIdx.x * 256 + threadIdx.x;
    if (idx >= (size_t)N * 256) return;
    int j = (int)(idx & 255);
    int n = (int)(idx >> 8);
    int h = j >> 6;
    float val = buf[idx] / den[(size_t)n * 4 + h] + bias[j];
    float sc  = g[j] * rsqrtf(v[j] + BN_EPS);
    val = (val - m[j]) * sc + b[j];
    buf[idx] = fmaxf(val, 0.f);                      // in place: acc -> h1
}

// ---------------------------------------------------------------------------
// Layer-2 edge kernels (H=1, D=64)
// ---------------------------------------------------------------------------
__global__ __launch_bounds__(256) void gat_e2_logits(
    const float* __restrict__ xl, const float* __restrict__ xr,
    const float* __restrict__ att, const int* __restrict__ ei,
    int E, int Etot, float* __restrict__ logit, unsigned* __restrict__ mx)
{
    int e = blockIdx.x * 8 + (threadIdx.x >> 5);
    if (e >= Etot) return;
    int lane = threadIdx.x & 31;
    int d = lane * 2;
    int src, dst;
    if (e < E) { src = ei[e]; dst = ei[E + e]; } else { src = dst = e - E; }
    float2 a = *(const float2*)(att + d);
    float2 l = *(const float2*)(xl + (size_t)src * 64 + d);
    float2 r = *(const float2*)(xr + (size_t)dst * 64 + d);
    float s = a.x * lrelu(l.x + r.x) + a.y * lrelu(l.y + r.y);
    s += __shfl_xor(s, 16); s += __shfl_xor(s, 8); s += __shfl_xor(s, 4);
    s += __shfl_xor(s, 2);  s += __shfl_xor(s, 1);
    if (lane == 0) { logit[e] = s; atomicMax(&mx[dst], fenc(s)); }
}

__global__ __launch_bounds__(256) void gat_e2_accum(
    const float* __restrict__ xl, const float* __restrict__ logit,
    const unsigned* __restrict__ mx, const int* __restrict__ ei,
    int E, int Etot, float* __restrict__ den, float* __restrict__ acc)
{
    int e = blockIdx.x * 8 + (threadIdx.x >> 5);
    if (e >= Etot) return;
    int lane = threadIdx.x & 31;
    int src, dst;
    if (e < E) { src = ei[e]; dst = ei[E + e]; } else { src = dst = e - E; }
    float a = __expf(logit[e] - fdec(mx[dst]));
    if (lane == 0) atomicAdd(&den[dst], a);
    int d = lane * 2;
    float2 xv = *(const float2*)(xl + (size_t)src * 64 + d);
    float* po = acc + (size_t)dst * 64 + d;
    atomicAdd(po + 0, a * xv.x);
    atomicAdd(po + 1, a * xv.y);
}

__global__ void gat_n2_post(
    float* __restrict__ buf, const float* __restrict__ den,
    const float* __restrict__ skip, const float* __restrict__ bias,
    const float* __restrict__ g, const float* __restrict__ b,
    const float* __restrict__ m, const float* __restrict__ v, int N)
{
    size_t idx = (size_t)blockIdx.x * 256 + threadIdx.x;
    if (idx >= (size_t)N * 64) return;
    int j = (int)(idx & 63);
    int n = (int)(idx >> 6);
    float val = buf[idx] / den[n] + bias[j];
    float sc  = g[j] * rsqrtf(v[j] + BN_EPS);
    val = (val - m[j]) * sc + b[j];
    buf[idx] = fmaxf(val, 0.f) + skip[idx];          // in place: acc -> h2+skip
}

__global__ __launch_bounds__(256) void gat_final_dot(
    const float* __restrict__ h, const float* __restrict__ Wo,
    const float* __restrict__ bo, float* __restrict__ out, int N)
{
    int n = blockIdx.x * 8 + (threadIdx.x >> 5);     // wave per node
    if (n >= N) return;
    int lane = threadIdx.x & 31;
    int d = lane * 2;
    float2 hv = *(const float2*)(h + (size_t)n * 64 + d);
    float2 wv = *(const float2*)(Wo + d);
    float s = hv.x * wv.x + hv.y * wv.y;
    s += __shfl_xor(s, 16); s += __shfl_xor(s, 8); s += __shfl_xor(s, 4);
    s += __shfl_xor(s, 2);  s += __shfl_xor(s, 1);
    if (lane == 0) out[n] = s + bo[0];
}

// ---------------------------------------------------------------------------
extern "C" void kernel_launch(void* const* d_in, const int* in_sizes, int n_in,
                              void* d_out, int out_size, void* d_ws, size_t ws_size,
                              hipStream_t stream)
{
    const float* x    = (const float*)d_in[0];
    const int*   ei   = (const int*)d_in[1];
    const float* Wl1  = (const float*)d_in[2];
    const float* bl1  = (const float*)d_in[3];
    const float* Wr1  = (const float*)d_in[4];
    const float* br1  = (const float*)d_in[5];
    const float* att1 = (const float*)d_in[6];
    const float* bias1= (const float*)d_in[7];
    const float* g1   = (const float*)d_in[8];
    const float* b1   = (const float*)d_in[9];
    const float* m1   = (const float*)d_in[10];
    const float* v1   = (const float*)d_in[11];
    const float* Wl2  = (const float*)d_in[12];
    const float* bl2  = (const float*)d_in[13];
    const float* Wr2  = (const float*)d_in[14];
    const float* br2  = (const float*)d_in[15];
    const float* att2 = (const float*)d_in[16];
    const float* bias2= (const float*)d_in[17];
    const float* g2   = (const float*)d_in[18];
    const float* b2   = (const float*)d_in[19];
    const float* m2   = (const float*)d_in[20];
    const float* v2   = (const float*)d_in[21];
    const float* Ws   = (const float*)d_in[22];
    const float* bs   = (const float*)d_in[23];
    const float* Wo   = (const float*)d_in[24];
    const float* bo   = (const float*)d_in[25];
    float* out = (float*)d_out;

    const int K1 = 160, F1 = 256, D = 64;
    const int N = in_sizes[0] / K1;       // 50000
    const int E = in_sizes[1] / 2;        // 800000
    const int Etot = E + N;               // with self-loops

    // workspace layout (floats)
    float* ws = (float*)d_ws;
    size_t o = 0;
    float* xl1   = ws + o; o += (size_t)N * F1;
    float* xr1   = ws + o; o += (size_t)N * F1;
    float* buf1  = ws + o; o += (size_t)N * F1;     // acc1 -> h1 (in place)
    float* xskip = ws + o; o += (size_t)N * D;
    float* xl2   = ws + o; o += (size_t)N * D;
    float* xr2   = ws + o; o += (size_t)N * D;
    float* buf2  = ws + o; o += (size_t)N * D;      // acc2 -> h2+skip (in place)
    unsigned* mx1 = (unsigned*)(ws + o); o += (size_t)N * 4;
    float* den1  = ws + o; o += (size_t)N * 4;
    unsigned* mx2 = (unsigned*)(ws + o); o += (size_t)N;
    float* den2  = ws + o; o += (size_t)N;
    float* lg1   = ws + o; o += (size_t)Etot * 4;
    float* lg2   = ws + o; o += (size_t)Etot;

    auto blks = [](size_t n, int t) { return (unsigned)((n + t - 1) / t); };

    // init accumulators / maxima (graph-safe, every call)
    fill_f32<<<blks((size_t)N * F1, 256), 256, 0, stream>>>(buf1, (size_t)N * F1, 0.f);
    fill_f32<<<blks((size_t)N * D,  256), 256, 0, stream>>>(buf2, (size_t)N * D, 0.f);
    fill_f32<<<blks((size_t)N * 4,  256), 256, 0, stream>>>(den1, (size_t)N * 4, 0.f);
    fill_f32<<<blks((size_t)N,      256), 256, 0, stream>>>(den2, (size_t)N, 0.f);
    fill_u32<<<blks((size_t)N * 4,  256), 256, 0, stream>>>(mx1, (size_t)N * 4, ENC_NEG_INF);
    fill_u32<<<blks((size_t)N,      256), 256, 0, stream>>>(mx2, (size_t)N, ENC_NEG_INF);

    // node transforms via LDS-staged WMMA-f32 GEMMs (64x64 tile per block)
    unsigned mtiles = (unsigned)((N + 63) / 64);
    dim3 gA(F1 / 64, mtiles);   // 256-wide outputs
    dim3 gB(D  / 64, mtiles);   // 64-wide outputs
    gemm_wmma_f32<<<gA, 128, 0, stream>>>(x, Wl1, bl1, xl1, N, K1, F1);
    gemm_wmma_f32<<<gA, 128, 0, stream>>>(x, Wr1, br1, xr1, N, K1, F1);
    gemm_wmma_f32<<<gB, 128, 0, stream>>>(x, Ws,  bs,  xskip, N, K1, D);

    // layer 1: logits + segment-max, then exp/scatter, then normalize+BN+ReLU
    gat_e1_logits<<<blks((size_t)Etot, 8), 256, 0, stream>>>(xl1, xr1, att1, ei, E, Etot, lg1, mx1);
    gat_e1_accum <<<blks((size_t)Etot, 4), 256, 0, stream>>>(xl1, lg1, mx1, ei, E, Etot, den1, buf1);
    gat_n1_post  <<<blks((size_t)N * F1, 256), 256, 0, stream>>>(buf1, den1, bias1, g1, b1, m1, v1, N);

    // layer 2 transforms (K=256)
    gemm_wmma_f32<<<gB, 128, 0, stream>>>(buf1, Wl2, bl2, xl2, N, F1, D);
    gemm_wmma_f32<<<gB, 128, 0, stream>>>(buf1, Wr2, br2, xr2, N, F1, D);

    gat_e2_logits<<<blks((size_t)Etot, 8), 256, 0, stream>>>(xl2, xr2, att2, ei, E, Etot, lg2, mx2);
    gat_e2_accum <<<blks((size_t)Etot, 8), 256, 0, stream>>>(xl2, lg2, mx2, ei, E, Etot, den2, buf2);
    gat_n2_post  <<<blks((size_t)N * D, 256), 256, 0, stream>>>(buf2, den2, xskip, bias2, g2, b2, m2, v2, N);

    gat_final_dot<<<blks((size_t)N, 8), 256, 0, stream>>>(buf2, Wo, bo, out, N);
}